// RNNModel_68461778698336
// MI455X (gfx1250) — compile-verified
//
#include <hip/hip_runtime.h>
#include <hip/hip_bf16.h>
#include <math.h>

#define INPUT      28
#define HIDDEN     16
#define CLASSES    35
#define SEQ        28
#define TILE_B     16
#define XROW       (SEQ * INPUT)   // 784 floats per batch row
#define CH_T       4               // timesteps per staged chunk
#define NCHUNK     (SEQ / CH_T)    // 7
#define CH_ROW     (CH_T * INPUT)  // 112 floats of one batch row per chunk
#define SRS        116             // padded LDS row stride (floats): 116%64=52, gcd(52,64)=4
                                   // -> conflict-free for 16-lane strided b64 reads
#define CH_F4      (TILE_B * CH_ROW / 4)   // 448 float4 per chunk
#define F4_PER_LANE (CH_F4 / 32)           // 14 async b128 per lane per chunk

typedef __attribute__((ext_vector_type(2))) float v2f;
typedef __attribute__((ext_vector_type(8))) float v8f;
typedef __attribute__((ext_vector_type(4))) int   v4i;

typedef __attribute__((address_space(1))) v4i* gv4i_p;  // global (prints as __device__)
typedef __attribute__((address_space(3))) v4i* lv4i_p;  // LDS    (prints as __shared__)

static __device__ inline v8f wmma4(v2f a, v2f b, v8f c) {
  // D = A(16x4 f32) * B(4x16 f32) + C(16x16 f32)
  return __builtin_amdgcn_wmma_f32_16x16x4_f32(false, a, false, b, (short)0, c,
                                               false, false);
}

// Swap lane halves 0..15 <-> 16..31 (wave32): ds_swizzle xor=0x10, and=0x1f.
static __device__ inline float halfswap(float x) {
  return __int_as_float(__builtin_amdgcn_ds_swizzle(__float_as_int(x), 0x401f));
}

struct B4 { v2f b[4]; };

// Convert 16x16 f32 matrix in WMMA C/D layout (reg r, lane l: row = r + 8*(l>>4),
// col = l&15) into four K=4 B-operand chunks: chunk c, reg v, lane (h,n) holds
// B[4c + v + 2h, n]. Rows 0..7 live in lane-half 0, rows 8..15 in half 1.
static __device__ inline B4 d_to_b(v8f g, int h) {
  B4 r;
#pragma unroll
  for (int c = 0; c < 2; ++c) {
#pragma unroll
    for (int v = 0; v < 2; ++v) {
      float direct = g[4 * c + v];               // k = 4c+v     (half-0 target)
      float sw     = halfswap(g[4 * c + v + 2]); // k = 4c+v+2   (half-1 target)
      r.b[c][v] = h ? sw : direct;
    }
  }
#pragma unroll
  for (int c = 2; c < 4; ++c) {
#pragma unroll
    for (int v = 0; v < 2; ++v) {
      float sw     = halfswap(g[4 * c + v - 8]); // k = 4c+v   >= 8 (half-0 target)
      float direct = g[4 * c + v - 6];           // k = 4c+v+2 >= 8 (half-1 target)
      r.b[c][v] = h ? direct : sw;
    }
  }
  return r;
}

static __device__ inline float fast_tanh(float x) {
#if __has_builtin(__builtin_amdgcn_tanhf)
  return __builtin_amdgcn_tanhf(x);   // v_tanh_f32 (CDNA5 trans op)
#else
  return tanhf(x);
#endif
}

// Async DMA: 16 bytes global -> LDS per lane, tracked by ASYNCcnt (no VGPR round-trip).
static __device__ inline void async_b128(const float* g, float* l) {
#if __has_builtin(__builtin_amdgcn_global_load_async_to_lds_b128)
  __builtin_amdgcn_global_load_async_to_lds_b128(
      (gv4i_p)(void*)(const_cast<float*>(g)),
      (lv4i_p)(void*)l, 0, 0);
#else
  unsigned loff = (unsigned)(unsigned long long)l;  // flat addr low 32 = LDS offset
  asm volatile("global_load_async_to_lds_b128 %0, %1, off"
               :: "v"(loff), "v"(g) : "memory");
#endif
}

#if __has_builtin(__builtin_amdgcn_s_wait_asynccnt)
#define WAIT_ASYNC(N) do { __builtin_amdgcn_s_wait_asynccnt(N); asm volatile("" ::: "memory"); } while (0)
#else
#define WAIT_ASYNC(N) asm volatile("s_wait_asynccnt %0" :: "i"(N) : "memory")
#endif

// Issue the 14 per-lane async b128 copies for chunk c into LDS buffer `buf`.
static __device__ inline void stage_chunk(const float* __restrict__ xbase,
                                          float* buf, int c, int lane) {
#pragma unroll
  for (int it = 0; it < F4_PER_LANE; ++it) {
    int idx  = it * 32 + lane;           // float4 index within chunk [0,448)
    int row  = idx / (CH_ROW / 4);       // 28 float4 per row-chunk
    int col4 = idx - row * (CH_ROW / 4);
    const float* g = xbase + row * XROW + c * CH_ROW + col4 * 4;
    float* l = buf + row * SRS + col4 * 4;
    async_b128(g, l);
  }
}

__global__ __launch_bounds__(32) void rnn2_wmma(
    const float* __restrict__ x,
    const float* __restrict__ W_ih0, const float* __restrict__ W_hh0,
    const float* __restrict__ b_ih0, const float* __restrict__ b_hh0,
    const float* __restrict__ W_ih1, const float* __restrict__ W_hh1,
    const float* __restrict__ b_ih1, const float* __restrict__ b_hh1,
    const float* __restrict__ fc_W,  const float* __restrict__ fc_b,
    float* __restrict__ out)
{
  __shared__ float xs[2][TILE_B * SRS];   // 2 x 7424 B double buffer
  const int lane = threadIdx.x;
  const int n = lane & 15;   // batch column within tile / weight output row
  const int h = lane >> 4;   // lane half
  const long b0 = (long)blockIdx.x * TILE_B;
  const float* __restrict__ xbase = x + b0 * XROW;

  // Kick off chunk 0 DMA immediately; weight loads below overlap it.
  stage_chunk(xbase, xs[0], 0, lane);

  // ---- Preload weights in WMMA A-operand layout: reg v, lane (h,m): W[m, 4j+v+2h] ----
  v2f a_ih0[7];
#pragma unroll
  for (int j = 0; j < 7; ++j) {
    int k0 = 4 * j + 2 * h;
    a_ih0[j][0] = W_ih0[n * INPUT + k0];
    a_ih0[j][1] = W_ih0[n * INPUT + k0 + 1];
  }
  v2f a_hh0[4], a_ih1[4], a_hh1[4];
#pragma unroll
  for (int j = 0; j < 4; ++j) {
    int k0 = 4 * j + 2 * h;
    a_hh0[j][0] = W_hh0[n * HIDDEN + k0]; a_hh0[j][1] = W_hh0[n * HIDDEN + k0 + 1];
    a_ih1[j][0] = W_ih1[n * HIDDEN + k0]; a_ih1[j][1] = W_ih1[n * HIDDEN + k0 + 1];
    a_hh1[j][0] = W_hh1[n * HIDDEN + k0]; a_hh1[j][1] = W_hh1[n * HIDDEN + k0 + 1];
  }
  v2f a_fc[3][4];
#pragma unroll
  for (int mt = 0; mt < 3; ++mt) {
    int row = mt * 16 + n;
#pragma unroll
    for (int j = 0; j < 4; ++j) {
      int k0 = 4 * j + 2 * h;
      float w0 = 0.f, w1 = 0.f;
      if (row < CLASSES) { w0 = fc_W[row * HIDDEN + k0]; w1 = fc_W[row * HIDDEN + k0 + 1]; }
      a_fc[mt][j][0] = w0; a_fc[mt][j][1] = w1;
    }
  }
  v8f bias0, bias1;
#pragma unroll
  for (int r = 0; r < 8; ++r) {
    int row = r + 8 * h;
    bias0[r] = b_ih0[row] + b_hh0[row];
    bias1[r] = b_ih1[row] + b_hh1[row];
  }

  // Hidden states kept transposed: G[hidden, batch] in C/D layout.
  v8f g1 = {}, g2 = {};

  for (int c = 0; c < NCHUNK; ++c) {
    // Prefetch next chunk into the other buffer, then wait for this chunk.
    // Async loads complete in order, so ASYNCcnt <= 14 means chunk c landed.
    if (c + 1 < NCHUNK) {
      stage_chunk(xbase, xs[(c + 1) & 1], c + 1, lane);
      WAIT_ASYNC(F4_PER_LANE);
    } else {
      WAIT_ASYNC(0);
    }
    const float* __restrict__ bufc = &xs[c & 1][n * SRS + 2 * h];

#pragma unroll
    for (int tt = 0; tt < CH_T; ++tt) {
      // ---- layer 0: pre = W_ih0 * x_t^T + W_hh0 * G1 + (b_ih0 + b_hh0) ----
      v8f acc = bias0;
      const float* __restrict__ xt = bufc + tt * INPUT;
#pragma unroll
      for (int j = 0; j < 7; ++j) {           // K = 28 = 7 x 4
        v2f bx = *(const v2f*)(xt + 4 * j);   // ds_load_b64, bank-conflict-free
        acc = wmma4(a_ih0[j], bx, acc);
      }
      B4 bg1 = d_to_b(g1, h);
#pragma unroll
      for (int k = 0; k < 4; ++k) acc = wmma4(a_hh0[k], bg1.b[k], acc);
      v8f g1n;
#pragma unroll
      for (int r = 0; r < 8; ++r) g1n[r] = fast_tanh(acc[r]);

      // ---- layer 1: pre = W_ih1 * G1n + W_hh1 * G2 + (b_ih1 + b_hh1) ----
      v8f acc2 = bias1;
      B4 bg1n = d_to_b(g1n, h);
#pragma unroll
      for (int k = 0; k < 4; ++k) acc2 = wmma4(a_ih1[k], bg1n.b[k], acc2);
      B4 bg2 = d_to_b(g2, h);
#pragma unroll
      for (int k = 0; k < 4; ++k) acc2 = wmma4(a_hh1[k], bg2.b[k], acc2);
#pragma unroll
      for (int r = 0; r < 8; ++r) g2[r] = fast_tanh(acc2[r]);
      g1 = g1n;
    }
  }

  // ---- classifier: out^T = fc_W * G2 + fc_b  (35 rows -> 3 padded M-tiles) ----
  B4 bg2f = d_to_b(g2, h);
#pragma unroll
  for (int mt = 0; mt < 3; ++mt) {
    v8f acc = {};
#pragma unroll
    for (int r = 0; r < 8; ++r) {
      int row = mt * 16 + r + 8 * h;
      acc[r] = (row < CLASSES) ? fc_b[row] : 0.f;
    }
#pragma unroll
    for (int k = 0; k < 4; ++k) acc = wmma4(a_fc[mt][k], bg2f.b[k], acc);
#pragma unroll
    for (int r = 0; r < 8; ++r) {
      int row = mt * 16 + r + 8 * h;
      if (row < CLASSES) out[(b0 + n) * CLASSES + row] = acc[r];
    }
  }
}

extern "C" void kernel_launch(void* const* d_in, const int* in_sizes, int n_in,
                              void* d_out, int out_size, void* d_ws, size_t ws_size,
                              hipStream_t stream) {
  (void)n_in; (void)out_size; (void)d_ws; (void)ws_size;
  const float* x     = (const float*)d_in[0];
  const float* W_ih0 = (const float*)d_in[1];
  const float* W_hh0 = (const float*)d_in[2];
  const float* b_ih0 = (const float*)d_in[3];
  const float* b_hh0 = (const float*)d_in[4];
  const float* W_ih1 = (const float*)d_in[5];
  const float* W_hh1 = (const float*)d_in[6];
  const float* b_ih1 = (const float*)d_in[7];
  const float* b_hh1 = (const float*)d_in[8];
  const float* fc_W  = (const float*)d_in[9];
  const float* fc_b  = (const float*)d_in[10];
  float* out = (float*)d_out;

  int batch = in_sizes[0] / (SEQ * INPUT);
  dim3 grid(batch / TILE_B), block(32);
  rnn2_wmma<<<grid, block, 0, stream>>>(x, W_ih0, W_hh0, b_ih0, b_hh0,
                                        W_ih1, W_hh1, b_ih1, b_hh1,
                                        fc_W, fc_b, out);
}